// MEGNet_State_876173328941
// MI455X (gfx1250) — compile-verified
//
#include <hip/hip_runtime.h>
#include <math.h>

typedef __attribute__((ext_vector_type(2))) float v2f;
typedef __attribute__((ext_vector_type(8))) float v8f;

#define NB 1024   // batch segments (B)
#define FD 32     // feature dim
#define HS 34     // padded LDS row stride (bank-conflict-free for 16-row tile reads)
#define MLP_T 512 // k_mlp block size: 16 waves -> double VGPR budget, no spills

// ---------------------------------------------------------------- zero ws
__global__ __launch_bounds__(256) void k_zero(float* __restrict__ p, int n) {
  int i = blockIdx.x * blockDim.x + threadIdx.x;
  if (i < n) p[i] = 0.f;
}

// ------------------------------------------------- edge scatter (LDS-privatized)
// one edge per wave-iteration: lane j accumulates dim j; 4-way unrolled to keep
// >=4 independent 128B streaming loads in flight per wave.
__global__ __launch_bounds__(1024) void k_edge_scatter(
    const long long* __restrict__ esrc,   // edge_index row 0 (first E entries)
    const long long* __restrict__ batch,
    const float* __restrict__ attr,
    float* __restrict__ gsum, float* __restrict__ gcnt, long long E)
{
  extern __shared__ float lds[];
  float* sSum = lds;            // NB*FD
  float* sCnt = lds + NB * FD;  // NB
  for (int i = threadIdx.x; i < NB * FD + NB; i += blockDim.x) lds[i] = 0.f;
  __syncthreads();

  const int lane = threadIdx.x & 31;
  const long long w  = ((long long)blockIdx.x * blockDim.x + threadIdx.x) >> 5;
  const long long nw = ((long long)gridDim.x * blockDim.x) >> 5;

  long long e = w;
  for (; e + 3 * nw < E; e += 4 * nw) {
    long long e0 = e, e1 = e + nw, e2 = e + 2 * nw, e3 = e + 3 * nw;
    long long ep = e + 8 * nw;
    if (ep < E) __builtin_prefetch(&attr[ep * FD + lane], 0, 0);  // global_prefetch_b8
    long long s0 = esrc[e0], s1 = esrc[e1], s2 = esrc[e2], s3 = esrc[e3];
    int b0 = (int)batch[s0], b1 = (int)batch[s1], b2 = (int)batch[s2], b3 = (int)batch[s3];
    float v0 = __builtin_nontemporal_load(&attr[e0 * FD + lane]);
    float v1 = __builtin_nontemporal_load(&attr[e1 * FD + lane]);
    float v2 = __builtin_nontemporal_load(&attr[e2 * FD + lane]);
    float v3 = __builtin_nontemporal_load(&attr[e3 * FD + lane]);
    atomicAdd(&sSum[b0 * FD + lane], v0);
    atomicAdd(&sSum[b1 * FD + lane], v1);
    atomicAdd(&sSum[b2 * FD + lane], v2);
    atomicAdd(&sSum[b3 * FD + lane], v3);
    if (lane == 0) {
      atomicAdd(&sCnt[b0], 1.f); atomicAdd(&sCnt[b1], 1.f);
      atomicAdd(&sCnt[b2], 1.f); atomicAdd(&sCnt[b3], 1.f);
    }
  }
  for (; e < E; e += nw) {
    long long s = esrc[e];
    int b = (int)batch[s];
    float v = __builtin_nontemporal_load(&attr[e * FD + lane]);
    atomicAdd(&sSum[b * FD + lane], v);
    if (lane == 0) atomicAdd(&sCnt[b], 1.f);
  }
  __syncthreads();
  for (int i = threadIdx.x; i < NB * FD; i += blockDim.x) atomicAdd(&gsum[i], sSum[i]);
  for (int i = threadIdx.x; i < NB; i += blockDim.x)      atomicAdd(&gcnt[i], sCnt[i]);
}

// ------------------------------------------------- node scatter (direct atomics)
__global__ __launch_bounds__(1024) void k_node_scatter(
    const float* __restrict__ x, const long long* __restrict__ batch,
    float* __restrict__ gsum, float* __restrict__ gcnt, int N)
{
  const int lane = threadIdx.x & 31;
  int w  = (blockIdx.x * blockDim.x + threadIdx.x) >> 5;
  int nw = (gridDim.x * blockDim.x) >> 5;
  int n = w;
  for (; n + 3 * nw < N; n += 4 * nw) {
    int n0 = n, n1 = n + nw, n2 = n + 2 * nw, n3 = n + 3 * nw;
    int b0 = (int)batch[n0], b1 = (int)batch[n1], b2 = (int)batch[n2], b3 = (int)batch[n3];
    float v0 = __builtin_nontemporal_load(&x[(long long)n0 * FD + lane]);
    float v1 = __builtin_nontemporal_load(&x[(long long)n1 * FD + lane]);
    float v2 = __builtin_nontemporal_load(&x[(long long)n2 * FD + lane]);
    float v3 = __builtin_nontemporal_load(&x[(long long)n3 * FD + lane]);
    atomicAdd(&gsum[b0 * FD + lane], v0);
    atomicAdd(&gsum[b1 * FD + lane], v1);
    atomicAdd(&gsum[b2 * FD + lane], v2);
    atomicAdd(&gsum[b3 * FD + lane], v3);
    if (lane == 0) {
      atomicAdd(&gcnt[b0], 1.f); atomicAdd(&gcnt[b1], 1.f);
      atomicAdd(&gcnt[b2], 1.f); atomicAdd(&gcnt[b3], 1.f);
    }
  }
  for (; n < N; n += nw) {
    int b = (int)batch[n];
    float v = __builtin_nontemporal_load(&x[(long long)n * FD + lane]);
    atomicAdd(&gsum[b * FD + lane], v);
    if (lane == 0) atomicAdd(&gcnt[b], 1.f);
  }
}

// ------------------------------------------------- fused MLP helpers
// One 16-row block at a time; two 16x16 col-tiles (c0,c1) share the A operand.
__device__ __forceinline__ void bias_stats_store(
    v8f& c0, v8f& c1, const float* __restrict__ bias, bool doRelu,
    float* sSum, float* sSq, float* __restrict__ H, int rb, int lane)
{
  const int l15 = lane & 15;
  const int lh8 = (lane >> 4) << 3;
  float bc0 = bias[l15], bc1 = bias[16 + l15];
  float s0 = 0.f, q0 = 0.f, s1 = 0.f, q1 = 0.f;
#pragma unroll
  for (int j = 0; j < 8; ++j) {
    float v0 = c0[j] + bc0;
    float v1 = c1[j] + bc1;
    if (doRelu) { v0 = fmaxf(v0, 0.f); v1 = fmaxf(v1, 0.f); }
    c0[j] = v0; c1[j] = v1;
    s0 += v0; q0 += v0 * v0;
    s1 += v1; q1 += v1 * v1;
  }
  atomicAdd(&sSum[l15], s0);       // ds_add_f32 (lanes n, n+16 share a column)
  atomicAdd(&sSq[l15], q0);
  atomicAdd(&sSum[16 + l15], s1);
  atomicAdd(&sSq[16 + l15], q1);
  // Store pre-norm values. Safe: this wave has finished all reads of these rows.
#pragma unroll
  for (int j = 0; j < 8; ++j) {
    int row = rb * 16 + j + lh8;   // C/D layout: VGPR j = rows j / j+8
    H[row * HS + l15]      = c0[j];
    H[row * HS + 16 + l15] = c1[j];
  }
}

__device__ __forceinline__ void gemm_block_lds(
    v8f& c0, v8f& c1, const float* __restrict__ H,
    const float* __restrict__ W, int rb, int lane)
{
  const int l15 = lane & 15, lh2 = (lane >> 4) << 1;
  const int row = rb * 16 + l15;
  c0 = (v8f){}; c1 = (v8f){};
#pragma unroll
  for (int k = 0; k < FD; k += 4) {
    int kk = k + lh2;                        // A 16x4 layout: lanes>=16 hold K+2,K+3
    v2f a, b0, b1;
    a.x = H[row * HS + kk];  a.y = H[row * HS + kk + 1];
    b0.x = W[kk * FD + l15];       b0.y = W[(kk + 1) * FD + l15];
    b1.x = W[kk * FD + 16 + l15];  b1.y = W[(kk + 1) * FD + 16 + l15];
    c0 = __builtin_amdgcn_wmma_f32_16x16x4_f32(false, a, false, b0, (short)0, c0, false, false);
    c1 = __builtin_amdgcn_wmma_f32_16x16x4_f32(false, a, false, b1, (short)0, c1, false, false);
  }
}

__global__ __launch_bounds__(MLP_T) void k_mlp(
    const float* __restrict__ ue_sum, const float* __restrict__ uv_sum,
    const float* __restrict__ ue_cnt, const float* __restrict__ uv_cnt,
    const float* __restrict__ state,
    const float* __restrict__ W1, const float* __restrict__ b1,
    const float* __restrict__ g1, const float* __restrict__ be1,
    const float* __restrict__ W2, const float* __restrict__ b2,
    const float* __restrict__ g2, const float* __restrict__ be2,
    const float* __restrict__ W3, const float* __restrict__ b3,
    const float* __restrict__ g3, const float* __restrict__ be3,
    float* __restrict__ out)
{
  extern __shared__ float lds[];
  float* H      = lds;               // NB*HS  (activations, padded stride)
  float* invE   = H + NB * HS;       // NB
  float* invV   = invE + NB;         // NB
  float* sSum   = invV + NB;         // FD
  float* sSq    = sSum + FD;         // FD
  float* sScale = sSq + FD;          // FD
  float* sShift = sScale + FD;       // FD

  const int tid = threadIdx.x;
  for (int i = tid; i < NB; i += MLP_T) {
    invE[i] = 1.0f / fmaxf(ue_cnt[i], 1.0f);
    invV[i] = 1.0f / fmaxf(uv_cnt[i], 1.0f);
  }
  if (tid < FD) { sSum[tid] = 0.f; sSq[tid] = 0.f; }
  __syncthreads();

  const int lane = tid & 31, wave = tid >> 5;      // 16 waves
  const int l15 = lane & 15, lh2 = (lane >> 4) << 1;

  // ---- layer 1: concat(u_e, u_v, state)[1024,96] @ W1[96,32], relu, BN ----
  for (int rb = wave * 4; rb < wave * 4 + 4; ++rb) {
    const int row = rb * 16 + l15;
    v8f c0 = (v8f){}, c1 = (v8f){};
    for (int k = 0; k < 96; k += 4) {
      int kk = k + lh2;                       // region uniform: k%4==0, splits at 32/64
      v2f a, b0, b1;
      if (kk < 32) {
        float iv = invE[row];
        a.x = ue_sum[row * FD + kk] * iv;
        a.y = ue_sum[row * FD + kk + 1] * iv;
      } else if (kk < 64) {
        float iv = invV[row];
        a.x = uv_sum[row * FD + (kk - 32)] * iv;
        a.y = uv_sum[row * FD + (kk - 31)] * iv;
      } else {
        a.x = state[row * FD + (kk - 64)];
        a.y = state[row * FD + (kk - 63)];
      }
      b0.x = W1[kk * FD + l15];       b0.y = W1[(kk + 1) * FD + l15];
      b1.x = W1[kk * FD + 16 + l15];  b1.y = W1[(kk + 1) * FD + 16 + l15];
      c0 = __builtin_amdgcn_wmma_f32_16x16x4_f32(false, a, false, b0, (short)0, c0, false, false);
      c1 = __builtin_amdgcn_wmma_f32_16x16x4_f32(false, a, false, b1, (short)0, c1, false, false);
    }
    bias_stats_store(c0, c1, b1, true, sSum, sSq, H, rb, lane);
  }

  // Three BN normalize passes + two more WMMA layers, sharing one H buffer.
  const float* Wl[2]  = { W2, W3 };
  const float* bl[2]  = { b2, b3 };
  const float* gl[3]  = { g1, g2, g3 };
  const float* bel[3] = { be1, be2, be3 };

  for (int layer = 0; layer < 3; ++layer) {
    __syncthreads();
    if (tid < FD) {                          // scale/shift from batch statistics
      const float r = 1.0f / (float)NB;
      float m  = sSum[tid] * r;
      float va = sSq[tid] * r - m * m;
      float is = 1.0f / sqrtf(va + 1e-5f);
      float sc = gl[layer][tid] * is;
      sScale[tid] = sc;
      sShift[tid] = bel[layer][tid] - m * sc;
      sSum[tid] = 0.f; sSq[tid] = 0.f;       // reset for next layer
    }
    __syncthreads();
    if (layer < 2) {
      // elementwise normalize in LDS
      for (int i = tid; i < NB * FD; i += MLP_T) {
        int row = i >> 5, col = i & (FD - 1);
        H[row * HS + col] = H[row * HS + col] * sScale[col] + sShift[col];
      }
      __syncthreads();
      // next GEMM: H @ W, relu only on layer index 0 here (i.e. layer 2 of net)
      for (int rb = wave * 4; rb < wave * 4 + 4; ++rb) {
        v8f c0, c1;
        gemm_block_lds(c0, c1, H, Wl[layer], rb, lane);
        bias_stats_store(c0, c1, bl[layer], layer == 0, sSum, sSq, H, rb, lane);
      }
    } else {
      // final normalize straight to global output (coalesced)
      for (int i = tid; i < NB * FD; i += MLP_T) {
        int row = i >> 5, col = i & (FD - 1);
        out[i] = H[row * HS + col] * sScale[col] + sShift[col];
      }
    }
  }
}

// ---------------------------------------------------------------- launcher
extern "C" void kernel_launch(void* const* d_in, const int* in_sizes, int n_in,
                              void* d_out, int out_size, void* d_ws, size_t ws_size,
                              hipStream_t stream)
{
  const float*     x     = (const float*)d_in[0];
  const long long* eidx  = (const long long*)d_in[1];   // int64 per reference
  const float*     eattr = (const float*)d_in[2];
  const float*     state = (const float*)d_in[3];
  const long long* batch = (const long long*)d_in[4];
  const float *W1 = (const float*)d_in[5],  *b1 = (const float*)d_in[6],
              *g1 = (const float*)d_in[7],  *be1 = (const float*)d_in[8];
  const float *W2 = (const float*)d_in[9],  *b2 = (const float*)d_in[10],
              *g2 = (const float*)d_in[11], *be2 = (const float*)d_in[12];
  const float *W3 = (const float*)d_in[13], *b3 = (const float*)d_in[14],
              *g3 = (const float*)d_in[15], *be3 = (const float*)d_in[16];
  float* out = (float*)d_out;

  const int       N = in_sizes[0] / FD;
  const long long E = (long long)in_sizes[1] / 2;

  float* ws = (float*)d_ws;
  float* ue_sum = ws;
  float* uv_sum = ws + NB * FD;
  float* ue_cnt = ws + 2 * NB * FD;
  float* uv_cnt = ws + 2 * NB * FD + NB;
  const int wsFloats = 2 * NB * FD + 2 * NB;

  k_zero<<<(wsFloats + 255) / 256, 256, 0, stream>>>(ws, wsFloats);

  const size_t edgeLds = (size_t)(NB * FD + NB) * sizeof(float);      // 132 KB
  k_edge_scatter<<<256, 1024, edgeLds, stream>>>(eidx, batch, eattr, ue_sum, ue_cnt, E);
  k_node_scatter<<<256, 1024, 0, stream>>>(x, batch, uv_sum, uv_cnt, N);

  const size_t mlpLds = (size_t)(NB * HS + 2 * NB + 4 * FD) * sizeof(float);  // ~148 KB
  k_mlp<<<1, MLP_T, mlpLds, stream>>>(ue_sum, uv_sum, ue_cnt, uv_cnt, state,
                                      W1, b1, g1, be1, W2, b2, g2, be2,
                                      W3, b3, g3, be3, out);
  (void)n_in; (void)out_size; (void)ws_size;
}